// GraphNeuralNetwork_80513456931397
// MI455X (gfx1250) — compile-verified
//
#include <hip/hip_runtime.h>
#include <hip/hip_bf16.h>

typedef __attribute__((ext_vector_type(2))) float v2f;
typedef __attribute__((ext_vector_type(8))) float v8f;

#define NEG_SLOPE 0.1f
#define BN_EPS 1e-5f

// ---------------------------------------------------------------------------
// Utility kernels
// ---------------------------------------------------------------------------
__global__ void zero_f_kernel(float* __restrict__ p, long n) {
  long i = (long)blockIdx.x * blockDim.x + threadIdx.x;
  long stride = (long)gridDim.x * blockDim.x;
  for (; i < n; i += stride) p[i] = 0.0f;
}

__global__ void zero_i_kernel(int* __restrict__ p, long n) {
  long i = (long)blockIdx.x * blockDim.x + threadIdx.x;
  long stride = (long)gridDim.x * blockDim.x;
  for (; i < n; i += stride) p[i] = 0;
}

// ---------------------------------------------------------------------------
// CSR build: in-degree count, two-level exclusive scan, slot fill
// ---------------------------------------------------------------------------
__global__ void cnt_kernel(const int* __restrict__ dst, int E, int* __restrict__ dcnt) {
  int i = blockIdx.x * blockDim.x + threadIdx.x;
  if (i < E) atomicAdd(&dcnt[dst[i]], 1);
}

__global__ void dinv_kernel(const int* __restrict__ dcnt, float* __restrict__ dinv, int N) {
  int i = blockIdx.x * blockDim.x + threadIdx.x;
  if (i < N) dinv[i] = rsqrtf((float)dcnt[i] + 1.0f);  // +1 = self loop
}

// Per-block inclusive scan (256-wide) + block totals
__global__ __launch_bounds__(256) void scan1_kernel(
    const int* __restrict__ dcnt, int* __restrict__ incl,
    int* __restrict__ bsum, int N) {
  __shared__ int s[256];
  int tid = threadIdx.x;
  int i = blockIdx.x * 256 + tid;
  int v = (i < N) ? dcnt[i] : 0;
  s[tid] = v;
  __syncthreads();
  for (int o = 1; o < 256; o <<= 1) {
    int t = (tid >= o) ? s[tid - o] : 0;
    __syncthreads();
    s[tid] += t;
    __syncthreads();
  }
  if (i < N) incl[i] = s[tid];
  if (tid == 255) bsum[blockIdx.x] = s[255];
}

// Single-block scan of up to 1024 block totals -> exclusive block offsets
__global__ __launch_bounds__(1024) void scan2_kernel(
    const int* __restrict__ bsum, int* __restrict__ boff, int nb) {
  __shared__ int s[1024];
  int tid = threadIdx.x;
  int v = (tid < nb) ? bsum[tid] : 0;
  s[tid] = v;
  __syncthreads();
  for (int o = 1; o < 1024; o <<= 1) {
    int t = (tid >= o) ? s[tid - o] : 0;
    __syncthreads();
    s[tid] += t;
    __syncthreads();
  }
  if (tid < nb) boff[tid] = s[tid] - v;  // exclusive
}

// offsets[i] = exclusive prefix of dcnt
__global__ void scan3_kernel(const int* __restrict__ incl, const int* __restrict__ dcnt,
                             const int* __restrict__ boff, int* __restrict__ offsets, int N) {
  int i = blockIdx.x * blockDim.x + threadIdx.x;
  if (i < N) offsets[i] = incl[i] - dcnt[i] + boff[i >> 8];
}

__global__ void fill_csr_kernel(const int* __restrict__ src, const int* __restrict__ dst,
                                const int* __restrict__ offsets, int* __restrict__ cursor,
                                int* __restrict__ csr_src, int E) {
  int e = blockIdx.x * blockDim.x + threadIdx.x;
  if (e < E) {
    int d = dst[e];
    int slot = offsets[d] + atomicAdd(&cursor[d], 1);
    csr_src[slot] = src[e];
  }
}

// ---------------------------------------------------------------------------
// fp32 WMMA GEMM:  out[M x 128] = A[M x 128] @ W[128 x 128] (+ bias)
// 8 waves / block; wave w computes the 16x16 tile at columns [16w, 16w+16).
// W is staged TRANSPOSED in LDS (two 64-k chunks) so each lane's B K-pair is
// contiguous -> one ds_load_b64 per WMMA. Rows padded to dodge bank conflicts.
// ---------------------------------------------------------------------------
__global__ __launch_bounds__(256) void gemm128_wmma_kernel(
    const float* __restrict__ A, const float* __restrict__ W,
    const float* __restrict__ bias, float* __restrict__ out, int M) {
  constexpr int AS = 132;  // padded row stride for A tile (16 x 128)
  constexpr int KS = 68;   // padded k-stride for transposed W chunk (128 x 64)
  __shared__ float Al[16 * AS];
  __shared__ float Wt[128 * KS];

  const int tid   = threadIdx.x;
  const int m0    = blockIdx.x * 16;
  const int wave  = tid >> 5;
  const int lane  = tid & 31;
  const int half  = lane >> 4;   // 0: K pair {0,1}; 1: K pair {2,3}
  const int l16   = lane & 15;
  const int cbase = wave * 16;

  // Stage A tile (16 x 128), clamp rows for the M-tail tile.
  for (int i = tid; i < 16 * 128; i += 256) {
    int r = i >> 7, c = i & 127;
    int row = m0 + r;
    if (row >= M) row = M - 1;
    Al[r * AS + c] = A[(long)row * 128 + c];
  }

  v8f acc = {};
  for (int kc = 0; kc < 128; kc += 64) {
    __syncthreads();  // protect previous Wt chunk / initial Al fill
    for (int i = tid; i < 64 * 128; i += 256) {
      int r = i >> 7;   // k within chunk
      int c = i & 127;  // output column
      Wt[c * KS + r] = W[(long)(kc + r) * 128 + c];
    }
    __syncthreads();
    #pragma unroll
    for (int k = 0; k < 64; k += 4) {
      int kk = k + 2 * half;  // even -> 8B-aligned LDS loads
      // A fragment: 16x4 fp32 (lanes 0-15: K={0,1}; lanes 16-31: K={2,3})
      v2f a = *(const v2f*)&Al[l16 * AS + kc + kk];
      // B fragment: 4x16 fp32, contiguous K pair from transposed W
      v2f b = *(const v2f*)&Wt[(cbase + l16) * KS + kk];
      acc = __builtin_amdgcn_wmma_f32_16x16x4_f32(
          false, a, false, b, (short)0, acc, false, false);
    }
  }

  // C/D layout: VGPR r -> rows {m0+r, m0+8+r} for lane halves
  #pragma unroll
  for (int r = 0; r < 8; ++r) {
    int mrow = m0 + r + 8 * half;
    if (mrow < M) {
      int col = cbase + l16;
      float v = acc[r];
      if (bias) v += bias[col];
      out[(long)mrow * 128 + col] = v;
    }
  }
}

// ---------------------------------------------------------------------------
// GCN aggregation, atomic-free: one wave per destination node.
// Lane owns 4 consecutive features. acc = h[d]*dinv^2 + b, then gather-sum
// over CSR in-edges (h is L2-resident, edge src made lane-uniform/scalar).
// ---------------------------------------------------------------------------
__global__ __launch_bounds__(256) void gcn_aggregate_kernel(
    const float* __restrict__ h, const int* __restrict__ csr_src,
    const int* __restrict__ offsets, const int* __restrict__ dcnt,
    const float* __restrict__ dinv, const float* __restrict__ b,
    float* __restrict__ agg, int N) {
  int node = blockIdx.x * 8 + (threadIdx.x >> 5);
  int lane = threadIdx.x & 31;
  if (node >= N) return;

  float di = dinv[node];
  float sc = di * di;
  int c4 = lane * 4;

  float4 hs = ((const float4*)(h + (long)node * 128))[lane];
  float4 acc;
  acc.x = hs.x * sc + b[c4 + 0];
  acc.y = hs.y * sc + b[c4 + 1];
  acc.z = hs.z * sc + b[c4 + 2];
  acc.w = hs.w * sc + b[c4 + 3];

  int beg = offsets[node];
  int cnt = dcnt[node];
  for (int j = 0; j < cnt; ++j) {
    int s = __builtin_amdgcn_readfirstlane(csr_src[beg + j]);
    float coef = dinv[s] * di;
    float4 v = ((const float4*)(h + (long)s * 128))[lane];
    acc.x += v.x * coef;
    acc.y += v.y * coef;
    acc.z += v.z * coef;
    acc.w += v.w * coef;
  }
  ((float4*)(agg + (long)node * 128))[lane] = acc;
}

// ---------------------------------------------------------------------------
// BatchNorm over axis 0 (feature columns of a [rows x 128] matrix)
// ---------------------------------------------------------------------------
__global__ __launch_bounds__(256) void bn_partial_kernel(
    const float* __restrict__ x, long n,
    float* __restrict__ gsum, float* __restrict__ gsq) {
  __shared__ float ls[128], lq[128];
  int tid = threadIdx.x;
  if (tid < 128) { ls[tid] = 0.0f; lq[tid] = 0.0f; }
  __syncthreads();
  long i = (long)blockIdx.x * blockDim.x + tid;
  long stride = (long)gridDim.x * blockDim.x;
  for (; i < n; i += stride) {
    float v = x[i];
    int c = (int)(i & 127);
    atomicAdd(&ls[c], v);
    atomicAdd(&lq[c], v * v);
  }
  __syncthreads();
  if (tid < 128) {
    atomicAdd(&gsum[tid], ls[tid]);
    atomicAdd(&gsq[tid], lq[tid]);
  }
}

__global__ void bn_finalize_kernel(const float* __restrict__ gsum,
                                   const float* __restrict__ gsq,
                                   const float* __restrict__ g,
                                   const float* __restrict__ be,
                                   float nrows,
                                   float* __restrict__ sa, float* __restrict__ sb) {
  int c = threadIdx.x;
  if (c < 128) {
    float mean = gsum[c] / nrows;
    float var  = gsq[c] / nrows - mean * mean;
    float istd = rsqrtf(var + BN_EPS);
    float a = istd * g[c];
    sa[c] = a;
    sb[c] = be[c] - mean * a;
  }
}

__global__ void bn_apply_leaky_kernel(float* __restrict__ x, long n,
                                      const float* __restrict__ sa,
                                      const float* __restrict__ sb) {
  long i = (long)blockIdx.x * blockDim.x + threadIdx.x;
  long stride = (long)gridDim.x * blockDim.x;
  for (; i < n; i += stride) {
    int c = (int)(i & 127);
    float v = x[i] * sa[c] + sb[c];
    x[i] = v > 0.0f ? v : NEG_SLOPE * v;
  }
}

// ---------------------------------------------------------------------------
// Global mean pool over graphs
// ---------------------------------------------------------------------------
__global__ void pool_sum_kernel(const float* __restrict__ h,
                                const int* __restrict__ batch,
                                float* __restrict__ pooled, long n) {
  long i = (long)blockIdx.x * blockDim.x + threadIdx.x;
  long stride = (long)gridDim.x * blockDim.x;
  for (; i < n; i += stride) {
    int row = (int)(i >> 7), c = (int)(i & 127);
    atomicAdd(&pooled[(long)batch[row] * 128 + c], h[i]);
  }
}

__global__ void pool_cnt_kernel(const int* __restrict__ batch,
                                float* __restrict__ cnt, int N) {
  int i = blockIdx.x * blockDim.x + threadIdx.x;
  if (i < N) atomicAdd(&cnt[batch[i]], 1.0f);
}

__global__ void pool_div_kernel(float* __restrict__ pooled,
                                const float* __restrict__ cnt, long n) {
  long i = (long)blockIdx.x * blockDim.x + threadIdx.x;
  long stride = (long)gridDim.x * blockDim.x;
  for (; i < n; i += stride)
    pooled[i] /= fmaxf(cnt[i >> 7], 1.0f);
}

// ---------------------------------------------------------------------------
// Host launcher
// ---------------------------------------------------------------------------
static inline int gsblocks(long n) {
  long b = (n + 255) / 256;
  if (b > 4096) b = 4096;
  if (b < 1) b = 1;
  return (int)b;
}

extern "C" void kernel_launch(void* const* d_in, const int* in_sizes, int n_in,
                              void* d_out, int out_size, void* d_ws, size_t ws_size,
                              hipStream_t stream) {
  const float* x     = (const float*)d_in[0];
  const int*   ei    = (const int*)d_in[1];
  const int*   batch = (const int*)d_in[2];
  // d_in[3] = n_graphs scalar (device); recover G from out_size instead.
  const float* W1  = (const float*)d_in[4];  const float* b1  = (const float*)d_in[5];
  const float* g1  = (const float*)d_in[6];  const float* be1 = (const float*)d_in[7];
  const float* W2  = (const float*)d_in[8];  const float* b2  = (const float*)d_in[9];
  const float* g2  = (const float*)d_in[10]; const float* be2 = (const float*)d_in[11];
  const float* W3  = (const float*)d_in[12]; const float* b3  = (const float*)d_in[13];
  const float* g3  = (const float*)d_in[14]; const float* be3 = (const float*)d_in[15];
  const float* Wf1 = (const float*)d_in[16]; const float* bf1 = (const float*)d_in[17];
  const float* g4  = (const float*)d_in[18]; const float* be4 = (const float*)d_in[19];
  const float* Wf2 = (const float*)d_in[20]; const float* bf2 = (const float*)d_in[21];

  const int N = in_sizes[0] / 128;
  const int E = in_sizes[1] / 2;
  const int G = out_size / 128;
  const int nb = (N + 255) / 256;  // scan blocks (<= 1024 supported)
  const int* src = ei;
  const int* dst = ei + E;

  // Workspace carve (4-byte elems, 64-elem alignment)
  char* wsb = (char*)d_ws;
  size_t off = 0;
  auto carve = [&](long n) { void* p = wsb + off; off += (size_t)((n + 63) & ~63L) * 4; return p; };
  int*   dcnt    = (int*)carve(N);
  int*   incl    = (int*)carve(N);
  int*   offsets = (int*)carve(N);
  int*   cursor  = (int*)carve(N);
  int*   bsum    = (int*)carve(1024);
  int*   boff    = (int*)carve(1024);
  int*   csr_src = (int*)carve(E);
  float* dinv    = (float*)carve(N);
  float* P       = (float*)carve((long)N * 128);   // gemm output h
  float* Q       = (float*)carve((long)N * 128);   // agg / normalized features
  float* gsum    = (float*)carve(128);
  float* gsq     = (float*)carve(128);
  float* sa      = (float*)carve(128);
  float* sb      = (float*)carve(128);
  float* pooled  = (float*)carve((long)G * 128);
  float* cnt     = (float*)carve(G);
  float* z1      = (float*)carve((long)G * 128);
  (void)ws_size;  // ~55 MB used

  const long NF = (long)N * 128;
  const long GF = (long)G * 128;

  // ---- CSR build (once; shared by all 3 layers) ----
  zero_i_kernel<<<gsblocks(N), 256, 0, stream>>>(dcnt, N);
  zero_i_kernel<<<gsblocks(N), 256, 0, stream>>>(cursor, N);
  cnt_kernel<<<(E + 255) / 256, 256, 0, stream>>>(dst, E, dcnt);
  dinv_kernel<<<(N + 255) / 256, 256, 0, stream>>>(dcnt, dinv, N);
  scan1_kernel<<<nb, 256, 0, stream>>>(dcnt, incl, bsum, N);
  scan2_kernel<<<1, 1024, 0, stream>>>(bsum, boff, nb);
  scan3_kernel<<<(N + 255) / 256, 256, 0, stream>>>(incl, dcnt, boff, offsets, N);
  fill_csr_kernel<<<(E + 255) / 256, 256, 0, stream>>>(src, dst, offsets, cursor, csr_src, E);

  const float* Ws[3]  = {W1, W2, W3};
  const float* bs[3]  = {b1, b2, b3};
  const float* gms[3] = {g1, g2, g3};
  const float* bes[3] = {be1, be2, be3};

  const float* feat = x;
  for (int l = 0; l < 3; ++l) {
    gemm128_wmma_kernel<<<(N + 15) / 16, 256, 0, stream>>>(feat, Ws[l], nullptr, P, N);
    gcn_aggregate_kernel<<<(N + 7) / 8, 256, 0, stream>>>(
        P, csr_src, offsets, dcnt, dinv, bs[l], Q, N);
    zero_f_kernel<<<1, 256, 0, stream>>>(gsum, 128);
    zero_f_kernel<<<1, 256, 0, stream>>>(gsq, 128);
    bn_partial_kernel<<<1024, 256, 0, stream>>>(Q, NF, gsum, gsq);
    bn_finalize_kernel<<<1, 128, 0, stream>>>(gsum, gsq, gms[l], bes[l], (float)N, sa, sb);
    bn_apply_leaky_kernel<<<gsblocks(NF), 256, 0, stream>>>(Q, NF, sa, sb);
    feat = Q;
  }

  // ---- Global mean pool ----
  zero_f_kernel<<<gsblocks(GF), 256, 0, stream>>>(pooled, GF);
  zero_f_kernel<<<gsblocks(G), 256, 0, stream>>>(cnt, G);
  pool_sum_kernel<<<gsblocks(NF), 256, 0, stream>>>(Q, batch, pooled, NF);
  pool_cnt_kernel<<<(N + 255) / 256, 256, 0, stream>>>(batch, cnt, N);
  pool_div_kernel<<<gsblocks(GF), 256, 0, stream>>>(pooled, cnt, GF);

  // ---- Dense head: z = leaky(bn(pooled @ Wf1 + bf1)); out = z @ Wf2 + bf2 ----
  gemm128_wmma_kernel<<<(G + 15) / 16, 256, 0, stream>>>(pooled, Wf1, bf1, z1, G);
  zero_f_kernel<<<1, 256, 0, stream>>>(gsum, 128);
  zero_f_kernel<<<1, 256, 0, stream>>>(gsq, 128);
  bn_partial_kernel<<<64, 256, 0, stream>>>(z1, GF, gsum, gsq);
  bn_finalize_kernel<<<1, 128, 0, stream>>>(gsum, gsq, g4, be4, (float)G, sa, sb);
  bn_apply_leaky_kernel<<<gsblocks(GF), 256, 0, stream>>>(z1, GF, sa, sb);
  gemm128_wmma_kernel<<<(G + 15) / 16, 256, 0, stream>>>(z1, Wf2, bf2, (float*)d_out, G);
}